// GNN_5136780886572
// MI455X (gfx1250) — compile-verified
//
#include <hip/hip_runtime.h>

#define N_NODES   100000
#define N_EDGES   1600000
#define E_TOT     (N_EDGES + N_NODES)
#define D_HID     128
#define NEG_SLOPE 0.2f
#define SCAN_B    256
#define ROWT      4   // 16-row tiles per wave in the WMMA GEMM

typedef __attribute__((ext_vector_type(2))) float v2f;
typedef __attribute__((ext_vector_type(8))) float v8f;

__device__ __forceinline__ float wave_max32(float v) {
#pragma unroll
  for (int m = 16; m > 0; m >>= 1) v = fmaxf(v, __shfl_xor(v, m, 32));
  return v;
}
__device__ __forceinline__ float wave_sum32(float v) {
#pragma unroll
  for (int m = 16; m > 0; m >>= 1) v += __shfl_xor(v, m, 32);
  return v;
}

// ---------------- CSR build (done once per launch; edges are layer-invariant) ---------

__global__ void k_init_counts(int* __restrict__ cnt) {
  int i = blockIdx.x * 256 + threadIdx.x;
  if (i < N_NODES) cnt[i] = 1;  // self loop contributes 1
}

__global__ void k_count_edges(const int* __restrict__ ei, int* __restrict__ cnt) {
  int e = blockIdx.x * 256 + threadIdx.x;
  if (e < N_EDGES) atomicAdd(&cnt[ei[N_EDGES + e]], 1);  // row 1 = dst
}

__global__ void k_scan_blocks(const int* __restrict__ cnt, int* __restrict__ rowptr,
                              int* __restrict__ bsums) {
  __shared__ int sm[SCAN_B];
  int tid = threadIdx.x;
  int gi = blockIdx.x * SCAN_B + tid;
  int v = (gi < N_NODES) ? cnt[gi] : 0;
  sm[tid] = v;
  __syncthreads();
  for (int off = 1; off < SCAN_B; off <<= 1) {
    int t = (tid >= off) ? sm[tid - off] : 0;
    __syncthreads();
    sm[tid] += t;
    __syncthreads();
  }
  int inc = sm[tid];
  if (gi < N_NODES) rowptr[gi] = inc - v;  // block-local exclusive
  if (tid == SCAN_B - 1) bsums[blockIdx.x] = inc;
}

__global__ void k_scan_sums(int* __restrict__ bsums, int nb) {
  __shared__ int sm[512];
  int tid = threadIdx.x;
  int v = (tid < nb) ? bsums[tid] : 0;
  sm[tid] = v;
  __syncthreads();
  for (int off = 1; off < 512; off <<= 1) {
    int t = (tid >= off) ? sm[tid - off] : 0;
    __syncthreads();
    sm[tid] += t;
    __syncthreads();
  }
  if (tid < nb) bsums[tid] = sm[tid] - v;  // exclusive block offsets
}

__global__ void k_add_offsets(int* __restrict__ rowptr, const int* __restrict__ bsums) {
  int gi = blockIdx.x * SCAN_B + threadIdx.x;
  if (gi < N_NODES) rowptr[gi] += bsums[blockIdx.x];
  if (gi == 0) rowptr[N_NODES] = E_TOT;
}

__global__ void k_copy_cursor(const int* __restrict__ rowptr, int* __restrict__ cur) {
  int i = blockIdx.x * 256 + threadIdx.x;
  if (i < N_NODES) cur[i] = rowptr[i];
}

__global__ void k_scatter_edges(const int* __restrict__ ei, int* __restrict__ cur,
                                int* __restrict__ srcs) {
  int e = blockIdx.x * 256 + threadIdx.x;
  if (e < N_EDGES) {
    int s = ei[e];
    int d = ei[N_EDGES + e];
    int pos = atomicAdd(&cur[d], 1);
    srcs[pos] = s;
  }
}

__global__ void k_scatter_self(int* __restrict__ cur, int* __restrict__ srcs) {
  int v = blockIdx.x * 256 + threadIdx.x;
  if (v < N_NODES) {
    int pos = atomicAdd(&cur[v], 1);
    srcs[pos] = v;
  }
}

// ---------------- H = X * W  (100000x128 @ 128x128) via fp32 WMMA ---------------------
// Each wave owns one 16-wide column tile and ROWT=4 row tiles (64 rows / block),
// so each B fragment (2 scalar W loads) feeds 4 WMMAs.
// Fragment mapping (lane l: lo = l&15, hi = l>>4):
//   A: a[0] = X[row+lo][k+2*hi], a[1] = X[row+lo][k+2*hi+1]
//   B: b[0] = W[k+2*hi][col+lo], b[1] = W[k+2*hi+1][col+lo]
//   D elem r -> H[rowtile + r + 8*hi][col + lo]
__global__ __launch_bounds__(256) void k_gemm128_wmma(const float* __restrict__ X,
                                                      const float* __restrict__ W,
                                                      float* __restrict__ H) {
  const int lane = threadIdx.x & 31;
  const int wave = threadIdx.x >> 5;  // 8 waves -> 8 column tiles of 16
  const int lo = lane & 15;
  const int hi = lane >> 4;
  const int rowbase0 = blockIdx.x * (16 * ROWT);
  const int colbase = wave * 16;

  const float* xr[ROWT];
#pragma unroll
  for (int rt = 0; rt < ROWT; ++rt) {
    int row = rowbase0 + rt * 16 + lo;
    if (row >= N_NODES) row = N_NODES - 1;  // tail clamp; stores guarded per-tile
    xr[rt] = X + (size_t)row * D_HID + 2 * hi;
  }
  const float* wc = W + (size_t)(2 * hi) * D_HID + colbase + lo;

  v8f c[ROWT];
#pragma unroll
  for (int rt = 0; rt < ROWT; ++rt)
#pragma unroll
    for (int r = 0; r < 8; ++r) c[rt][r] = 0.0f;

#pragma unroll 4
  for (int k = 0; k < D_HID; k += 4) {
    v2f b;
    b[0] = wc[(size_t)k * D_HID];
    b[1] = wc[(size_t)(k + 1) * D_HID];
#pragma unroll
    for (int rt = 0; rt < ROWT; ++rt) {
      float2 av = *(const float2*)(xr[rt] + k);
      v2f a;
      a[0] = av.x;
      a[1] = av.y;
      c[rt] = __builtin_amdgcn_wmma_f32_16x16x4_f32(false, a, false, b, (short)0,
                                                    c[rt], false, false);
    }
  }

#pragma unroll
  for (int rt = 0; rt < ROWT; ++rt) {
    int tilebase = rowbase0 + rt * 16;  // N_NODES % 16 == 0: tile all-valid or all-invalid
    if (tilebase < N_NODES) {
      float* hb = H + (size_t)(tilebase + 8 * hi) * D_HID + colbase + lo;
#pragma unroll
      for (int r = 0; r < 8; ++r) hb[(size_t)r * D_HID] = c[rt][r];
    }
  }
}

// ---------------- a_s = H @ a_src, a_d = H @ a_dst (one wave per node) ----------------

template <int D>
__global__ __launch_bounds__(256) void k_attn_vec(const float* __restrict__ H,
                                                  const float* __restrict__ vs,
                                                  const float* __restrict__ vd,
                                                  float* __restrict__ a_s,
                                                  float* __restrict__ a_d) {
  int node = blockIdx.x * 8 + (threadIdx.x >> 5);
  int lane = threadIdx.x & 31;
  if (node >= N_NODES) return;
  const float* h = H + (size_t)node * D;
  float s = 0.f, t = 0.f;
  if constexpr (D == 128) {
    float4 hv = *(const float4*)(h + lane * 4);
    float4 s4 = *(const float4*)(vs + lane * 4);
    float4 d4 = *(const float4*)(vd + lane * 4);
    s = hv.x * s4.x + hv.y * s4.y + hv.z * s4.z + hv.w * s4.w;
    t = hv.x * d4.x + hv.y * d4.y + hv.z * d4.z + hv.w * d4.w;
  } else {
    for (int d = lane; d < D; d += 32) {
      float hv = h[d];
      s += hv * vs[d];
      t += hv * vd[d];
    }
  }
  s = wave_sum32(s);
  t = wave_sum32(t);
  if (lane == 0) {
    a_s[node] = s;
    a_d[node] = t;
  }
}

// ---------------- per-dst segment softmax + weighted aggregation (no atomics) ---------

template <int D, bool RELU>
__global__ __launch_bounds__(256) void k_aggregate(const float* __restrict__ H,
                                                   const float* __restrict__ a_s,
                                                   const float* __restrict__ a_d,
                                                   const int* __restrict__ rowptr,
                                                   const int* __restrict__ srcs,
                                                   const float* __restrict__ bias,
                                                   float* __restrict__ OUT) {
  int node = blockIdx.x * 8 + (threadIdx.x >> 5);
  int lane = threadIdx.x & 31;
  if (node >= N_NODES) return;
  int beg = rowptr[node], end = rowptr[node + 1];
  float ad = a_d[node];

  // pass 1: segment max (lanes stride over edges)
  float m = -1e30f;
  for (int i = beg + lane; i < end; i += 32) {
    float e = a_s[srcs[i]] + ad;
    e = e > 0.f ? e : NEG_SLOPE * e;
    m = fmaxf(m, e);
  }
  m = wave_max32(m);

  // pass 2: sum of exp
  float z = 0.f;
  for (int i = beg + lane; i < end; i += 32) {
    float e = a_s[srcs[i]] + ad;
    e = e > 0.f ? e : NEG_SLOPE * e;
    z += __expf(e - m);
  }
  z = wave_sum32(z);
  float inv = 1.0f / (z + 1e-16f);

  // pass 3: weighted feature sum; each lane owns a contiguous float4 of dims
  if constexpr (D == 128) {
    float4 acc = make_float4(0.f, 0.f, 0.f, 0.f);
    for (int i = beg; i < end; ++i) {
      int s = srcs[i];
      float e = a_s[s] + ad;
      e = e > 0.f ? e : NEG_SLOPE * e;
      float al = __expf(e - m) * inv;
      float4 hv = *(const float4*)(H + (size_t)s * D + lane * 4);  // one b128/lane
      acc.x += al * hv.x;
      acc.y += al * hv.y;
      acc.z += al * hv.z;
      acc.w += al * hv.w;
    }
    float4 bb = *(const float4*)(bias + lane * 4);
    float4 o;
    o.x = acc.x + bb.x;
    o.y = acc.y + bb.y;
    o.z = acc.z + bb.z;
    o.w = acc.w + bb.w;
    if (RELU) {
      o.x = fmaxf(o.x, 0.f);
      o.y = fmaxf(o.y, 0.f);
      o.z = fmaxf(o.z, 0.f);
      o.w = fmaxf(o.w, 0.f);
    }
    *(float4*)(OUT + (size_t)node * D + lane * 4) = o;
  } else {
    constexpr int PER = (D + 31) / 32;
    float acc[PER];
#pragma unroll
    for (int j = 0; j < PER; ++j) acc[j] = 0.f;
    for (int i = beg; i < end; ++i) {
      int s = srcs[i];
      float e = a_s[s] + ad;
      e = e > 0.f ? e : NEG_SLOPE * e;
      float al = __expf(e - m) * inv;
      const float* hr = H + (size_t)s * D;
#pragma unroll
      for (int j = 0; j < PER; ++j) {
        int d = lane + 32 * j;
        if (d < D) acc[j] += al * hr[d];
      }
    }
    float* outr = OUT + (size_t)node * D;
#pragma unroll
    for (int j = 0; j < PER; ++j) {
      int d = lane + 32 * j;
      if (d < D) {
        float v = acc[j] + bias[d];
        if (RELU) v = fmaxf(v, 0.f);
        outr[d] = v;
      }
    }
  }
}

// ---------------- final projection 128 -> 2 (one wave per node) -----------------------

__global__ __launch_bounds__(256) void k_gemm_out2(const float* __restrict__ X,
                                                   const float* __restrict__ W,  // [128][2]
                                                   float* __restrict__ H2) {
  int node = blockIdx.x * 8 + (threadIdx.x >> 5);
  int lane = threadIdx.x & 31;
  if (node >= N_NODES) return;
  const float4 xv = *(const float4*)(X + (size_t)node * 128 + lane * 4);
  int k = lane * 4;
  float s0 = xv.x * W[k * 2 + 0] + xv.y * W[(k + 1) * 2 + 0] +
             xv.z * W[(k + 2) * 2 + 0] + xv.w * W[(k + 3) * 2 + 0];
  float s1 = xv.x * W[k * 2 + 1] + xv.y * W[(k + 1) * 2 + 1] +
             xv.z * W[(k + 2) * 2 + 1] + xv.w * W[(k + 3) * 2 + 1];
  s0 = wave_sum32(s0);
  s1 = wave_sum32(s1);
  if (lane == 0) {
    H2[(size_t)node * 2 + 0] = s0;
    H2[(size_t)node * 2 + 1] = s1;
  }
}

// --------------------------------------------------------------------------------------

extern "C" void kernel_launch(void* const* d_in, const int* in_sizes, int n_in,
                              void* d_out, int out_size, void* d_ws, size_t ws_size,
                              hipStream_t stream) {
  const float* X0 = (const float*)d_in[0];
  const int* EI = (const int*)d_in[1];  // int64 in source but JAX x64 off -> int32
  // d_in[2] = edge_attr, unused (GATConv without edge_dim)
  const float *Wl[8], *Asl[8], *Adl[8], *Bl[8];
  for (int i = 0; i < 8; ++i) {
    Wl[i] = (const float*)d_in[3 + 4 * i + 0];
    Asl[i] = (const float*)d_in[3 + 4 * i + 1];
    Adl[i] = (const float*)d_in[3 + 4 * i + 2];
    Bl[i] = (const float*)d_in[3 + 4 * i + 3];
  }

  char* ws = (char*)d_ws;
  size_t off = 0;
  auto carve = [&](size_t bytes) -> void* {
    void* p = ws + off;
    off += (bytes + 255) & ~(size_t)255;
    return p;
  };
  float* Hbuf = (float*)carve((size_t)N_NODES * D_HID * 4);
  float* F = (float*)carve((size_t)N_NODES * D_HID * 4);
  float* a_s = (float*)carve((size_t)N_NODES * 4);
  float* a_d = (float*)carve((size_t)N_NODES * 4);
  int* rowptr = (int*)carve((size_t)(N_NODES + 1) * 4);
  int* cnt = (int*)carve((size_t)N_NODES * 4);  // also reused as scatter cursor
  int* bsums = (int*)carve(4096);
  int* srcs = (int*)carve((size_t)E_TOT * 4);
  float* H2 = (float*)carve((size_t)N_NODES * 2 * 4);

  const int nbN = (N_NODES + 255) / 256;  // 391 (fits scan_sums' 512-thread block)
  const int nbE = (N_EDGES + 255) / 256;

  // CSR by destination (includes self loops), built once, reused by all 8 layers
  k_init_counts<<<nbN, 256, 0, stream>>>(cnt);
  k_count_edges<<<nbE, 256, 0, stream>>>(EI, cnt);
  k_scan_blocks<<<nbN, 256, 0, stream>>>(cnt, rowptr, bsums);
  k_scan_sums<<<1, 512, 0, stream>>>(bsums, nbN);
  k_add_offsets<<<nbN, 256, 0, stream>>>(rowptr, bsums);
  k_copy_cursor<<<nbN, 256, 0, stream>>>(rowptr, cnt);
  k_scatter_edges<<<nbE, 256, 0, stream>>>(EI, cnt, srcs);
  k_scatter_self<<<nbN, 256, 0, stream>>>(cnt, srcs);

  const int nGemm = (N_NODES + 16 * ROWT - 1) / (16 * ROWT);  // 1563
  const int nAgg = (N_NODES + 7) / 8;                         // 12500

  const float* cur = X0;
  for (int l = 0; l < 7; ++l) {
    k_gemm128_wmma<<<nGemm, 256, 0, stream>>>(cur, Wl[l], Hbuf);
    k_attn_vec<D_HID><<<nAgg, 256, 0, stream>>>(Hbuf, Asl[l], Adl[l], a_s, a_d);
    k_aggregate<D_HID, true><<<nAgg, 256, 0, stream>>>(Hbuf, a_s, a_d, rowptr, srcs,
                                                       Bl[l], F);
    cur = F;
  }
  // layer 8: 128 -> 2, no relu, output straight to d_out
  k_gemm_out2<<<nAgg, 256, 0, stream>>>(cur, Wl[7], H2);
  k_attn_vec<2><<<nAgg, 256, 0, stream>>>(H2, Asl[7], Adl[7], a_s, a_d);
  k_aggregate<2, false><<<nAgg, 256, 0, stream>>>(H2, a_s, a_d, rowptr, srcs, Bl[7],
                                                  (float*)d_out);
}